// SWPTLayer_66194035966103
// MI455X (gfx1250) — compile-verified
//
#include <hip/hip_runtime.h>

typedef float v2f __attribute__((ext_vector_type(2)));
typedef float v8f __attribute__((ext_vector_type(8)));

// 2^-1.5 = (1/sqrt(2))^3 : total Haar scaling over 3 levels
#define S3 0.35355339059327376220f

// out[row, n*1024 + i] = S3 * sum_t (-1)^{popcount(bitrev3(n) & t)} x[row, 8i + t]
// Implemented as D(16x16) = diag(H8,H8) * X(16x16) via 4 chained
// V_WMMA_F32_16X16X4_F32 (K = 0..15). Column j of X = 16 consecutive inputs;
// top half of the 16-row diag handles even groups, bottom half odd groups.
__global__ __launch_bounds__(256) void swpt_haar3_wmma(const float* __restrict__ x,
                                                       float* __restrict__ out) {
  const int tid  = threadIdx.x;
  const int lane = tid & 31;
  const int wave = tid >> 5;        // 0..7
  const int l    = lane & 15;
  const int half = lane >> 4;       // 0: K even slot / N 0..15, 1: K odd slot

  const int row = blockIdx.x;       // 0..4095 (= 64*64 rows of 8192)
  const float* __restrict__ in_row  = x   + (size_t)row * 8192u;
  float*       __restrict__ out_row = out + (size_t)row * 8192u;

  // ---- constant A = diag(H,H)*S3 in WMMA 16x4 f32 A layout ----
  // lane -> M = lane&15 ; vgpr0 -> K = 4c + 2*half ; vgpr1 -> K = 4c + 2*half + 1
  const int M  = l;
  const int n_ = M & 7;
  // bit-reverse 3-bit node index: level-1 sign bit pairs with t bit0
  const int br = ((n_ & 1) << 2) | (n_ & 2) | ((n_ >> 2) & 1);
  v2f a[4];
#pragma unroll
  for (int c = 0; c < 4; ++c) {
#pragma unroll
    for (int e = 0; e < 2; ++e) {
      const int k = 4 * c + 2 * half + e;         // global K index 0..15
      float v = 0.0f;
      if ((M >> 3) == (k >> 3)) {                 // block-diagonal
        v = (__builtin_popcount(br & (k & 7)) & 1) ? -S3 : S3;
      }
      a[c][e] = v;
    }
  }

  // ---- 4 tiles per wave; tile T covers inputs [256*T, 256*T+256) ----
#pragma unroll
  for (int tt = 0; tt < 4; ++tt) {
    const int T = wave * 4 + tt;                  // 0..31
    // B 4x16 f32 layout: lane L<16 -> (K=4c, 4c+1) at x[16L+4c+{0,1}];
    //                    lane L>=16 -> (K=4c+2, 4c+3) at x[16(L-16)+4c+{2,3}]
    const float* __restrict__ bi = in_row + 256 * T + 16 * l + 2 * half;
    v2f b[4];
#pragma unroll
    for (int c = 0; c < 4; ++c) {
      b[c] = *(const v2f*)(bi + 4 * c);           // 8B-aligned float2
    }

    v8f acc = {};
#pragma unroll
    for (int c = 0; c < 4; ++c) {
      // (neg_a, A, neg_b, B, c_mod, C, reuse_a, reuse_b)
      acc = __builtin_amdgcn_wmma_f32_16x16x4_f32(false, a[c], false, b[c],
                                                  (short)0, acc, false, false);
    }

    // D layout: VGPR r, lanes 0-15 -> out[r*1024 + 32T + 2l],
    //                  lanes 16-31 -> out[r*1024 + 32T + 2(l)+1]
    float* __restrict__ po = out_row + 32 * T + 2 * l + half;
#pragma unroll
    for (int r = 0; r < 8; ++r) {
      po[r * 1024] = acc[r];                      // 8 coalesced b32 stores
    }
  }
}

extern "C" void kernel_launch(void* const* d_in, const int* in_sizes, int n_in,
                              void* d_out, int out_size, void* d_ws, size_t ws_size,
                              hipStream_t stream) {
  const float* x = (const float*)d_in[0];
  float* out = (float*)d_out;
  const int total = in_sizes[0];        // 64*64*8192
  const int rows  = total / 8192;       // 4096
  swpt_haar3_wmma<<<rows, 256, 0, stream>>>(x, out);
}